// Add_31318901522623
// MI455X (gfx1250) — compile-verified
//
#include <hip/hip_runtime.h>
#include <cstdint>

// Geometry: (B, L=64, K=2) f32 digit distributions; serial carry scan over L.
#define ELEMS 64                 // batch elements per block == blockDim.x (2 waves)
#define LBITS 64
#define KD    2
#define TB    16                 // bits per LDS tile
#define NT    (LBITS / TB)       // 4 tiles
#define PAIRS (TB / 2)           // 8 bit-pairs (one b128 each) per tile
#define ROWB  (ELEMS * 16)       // 1024 B per bit-pair row in LDS
#define OPB   (PAIRS * ROWB)     // 8 KB per operand tile
// total static LDS: 2 buffers * 2 operands * 8KB = 32 KB

typedef __attribute__((ext_vector_type(4))) float v4f;

__device__ __forceinline__ void bit_step(float a0, float a1, float b0, float b1,
                                         float& c0, float& c1, float& r0, float& r1) {
  float p00 = a0 * b0, p01 = a0 * b1, p10 = a1 * b0, p11 = a1 * b1;
  float se = p00 + p11, so = p01 + p10;          // even / odd digit-sum mass
  r0 = fmaf(se, c0, so * c1);                    // result digit 0
  r1 = fmaf(so, c0, se * c1);                    // result digit 1
  float cs = c0 + c1;
  float nc0 = fmaf(p00, cs, so * c0);            // carry-out 0
  float nc1 = fmaf(p11, cs, so * c1);            // carry-out 1
  c0 = nc0; c1 = nc1;
}

__global__ __launch_bounds__(ELEMS) void ripple_kernel(const float* __restrict__ op1,
                                                       const float* __restrict__ op2,
                                                       float* __restrict__ out) {
  __shared__ __align__(16) unsigned char smem[2 * 2 * OPB];
  // Flat LDS address truncates to the workgroup-relative LDS byte offset
  // async-to-LDS VDST wants that offset.
  const uint32_t lds_base = (uint32_t)(uintptr_t)&smem[0];

  const int tid = threadIdx.x;
  const uint64_t elem0 = (uint64_t)blockIdx.x * ELEMS;
  const uint64_t g1 = (uint64_t)(uintptr_t)(op1 + elem0 * (LBITS * KD));
  const uint64_t g2 = (uint64_t)(uintptr_t)(op2 + elem0 * (LBITS * KD));
  float* o = out + elem0 * (LBITS * KD) + (uint64_t)tid * (LBITS * KD);

  // Issue one tile's async global->LDS copies (transposing to bitpair-major).
  auto issue_tile = [&](int bi, int l0) {
    uint32_t lbase1 = lds_base + (uint32_t)(bi * 2 * OPB);
    uint32_t lbase2 = lbase1 + OPB;
#pragma unroll
    for (int k = 0; k < PAIRS; ++k) {
      int p = k * ELEMS + tid;
      int e = p >> 3;                 // element within block     (PAIRS == 8)
      int r = p & 7;                  // bit-pair index
      // global: element row is 512B; tile chunk [l0*8, l0*8+128) is a full line
      uint64_t goff = (uint64_t)e * (LBITS * KD * 4) +
                      (uint32_t)(l0 * KD * 4) + (uint32_t)(r * 16);
      uint32_t loff = (uint32_t)(r * ROWB + e * 16);
      uint64_t ga1 = g1 + goff;
      uint64_t ga2 = g2 + goff;
      asm volatile("global_load_async_to_lds_b128 %0, %1, off"
                   :: "v"(lbase1 + loff), "v"(ga1) : "memory");
      asm volatile("global_load_async_to_lds_b128 %0, %1, off"
                   :: "v"(lbase2 + loff), "v"(ga2) : "memory");
    }
  };

  issue_tile(0, 0);                       // prologue: tile 0 -> buffer 0

  float c0 = 1.0f, c1 = 0.0f;             // carry-in dist = [1, 0]
#pragma unroll
  for (int t = 0; t < NT; ++t) {
    if (t + 1 < NT) {
      issue_tile((t + 1) & 1, (t + 1) * TB);     // +16 async instr this wave
      asm volatile("s_wait_asynccnt 16" ::: "memory");  // tile t complete
    } else {
      asm volatile("s_wait_asynccnt 0" ::: "memory");
    }
    __syncthreads();                      // all waves' tile-t data visible

    const unsigned char* s1 = &smem[(t & 1) * 2 * OPB];
    const unsigned char* s2 = s1 + OPB;
    v4f res[PAIRS];
#pragma unroll
    for (int r = 0; r < PAIRS; ++r) {     // bits l0+2r, l0+2r+1 in scan order
      v4f a = *(const v4f*)(s1 + r * ROWB + tid * 16);
      v4f b = *(const v4f*)(s2 + r * ROWB + tid * 16);
      float r0, r1, r2, r3;
      bit_step(a.x, a.y, b.x, b.y, c0, c1, r0, r1);
      bit_step(a.z, a.w, b.z, b.w, c0, c1, r2, r3);
      res[r] = (v4f){r0, r1, r2, r3};
    }
#pragma unroll
    for (int r = 0; r < PAIRS; ++r) {     // 8 x b128 NT stores = one full 128B line
      __builtin_nontemporal_store(res[r], (v4f*)(o + t * (TB * KD) + r * 4));
    }
    __syncthreads();                      // buffer (t&1) reusable for tile t+2
  }
}

// Scalar fallback for a non-multiple-of-ELEMS batch tail (not hit for B=262144).
__global__ void ripple_tail(const float* __restrict__ op1, const float* __restrict__ op2,
                            float* __restrict__ out, long long start, long long total) {
  long long i = start + (long long)blockIdx.x * blockDim.x + threadIdx.x;
  if (i >= total) return;
  const float* a = op1 + i * (LBITS * KD);
  const float* b = op2 + i * (LBITS * KD);
  float* o = out + i * (LBITS * KD);
  float c0 = 1.0f, c1 = 0.0f;
  for (int l = 0; l < LBITS; ++l) {
    float r0, r1;
    bit_step(a[l * 2], a[l * 2 + 1], b[l * 2], b[l * 2 + 1], c0, c1, r0, r1);
    o[l * 2] = r0;
    o[l * 2 + 1] = r1;
  }
}

extern "C" void kernel_launch(void* const* d_in, const int* in_sizes, int n_in,
                              void* d_out, int out_size, void* d_ws, size_t ws_size,
                              hipStream_t stream) {
  const float* op1 = (const float*)d_in[0];
  const float* op2 = (const float*)d_in[1];
  float* out = (float*)d_out;
  long long elems = (long long)in_sizes[0] / (LBITS * KD);   // 262144
  long long full = elems / ELEMS;                            // 4096 blocks
  if (full > 0)
    ripple_kernel<<<(unsigned)full, ELEMS, 0, stream>>>(op1, op2, out);
  long long done = full * ELEMS;
  if (done < elems) {
    long long rem = elems - done;
    ripple_tail<<<(unsigned)((rem + 63) / 64), 64, 0, stream>>>(op1, op2, out, done, elems);
  }
}